// _SCELoss_24206435680478
// MI455X (gfx1250) — compile-verified
//
#include <hip/hip_runtime.h>
#include <hip/hip_bf16.h>

#define N_CLASSES_ 1000
#define N_BINS_    15
#define NSEG_      (N_CLASSES_ * N_BINS_)
#define ROW_F4_    250   // 1000 f32 = 250 float4 = 4000 bytes per row

// ---- gfx1250 async global->LDS path (guarded so the TU always compiles) ----
#if defined(__HIP_DEVICE_COMPILE__) && defined(__gfx1250__)
#  if __has_builtin(__builtin_amdgcn_global_load_async_to_lds_b128) && \
      __has_builtin(__builtin_amdgcn_s_wait_asynccnt)
#    define USE_ASYNC_LDS 1
#  endif
#endif
#ifndef USE_ASYNC_LDS
#  define USE_ASYNC_LDS 0
#endif

#if USE_ASYNC_LDS
typedef int v4i __attribute__((ext_vector_type(4)));
typedef __attribute__((address_space(1))) v4i* gas1_t;  // global int4*
typedef __attribute__((address_space(3))) v4i* las3_t;  // LDS int4*

// Issue 8 async b128 transfers covering one 4000-byte row (250 float4).
// Lane l owns float4 chunks l + 32*i; byte offset = 16*l + 512*i.
// ISA: INST_OFFSET is added to BOTH the global and LDS per-lane addresses,
// so one base address pair + immediate offsets covers the whole row.
__device__ __forceinline__ void async_prefetch_row(const float* rowp, float4* buf, int lane) {
    const char* gb = (const char*)rowp + (size_t)lane * 16;
    char*       lb = (char*)buf        + (size_t)lane * 16;
    gas1_t g = (gas1_t)(void*)gb;
    las3_t l = (las3_t)(void*)lb;
    __builtin_amdgcn_global_load_async_to_lds_b128(g, l,    0, 0);
    __builtin_amdgcn_global_load_async_to_lds_b128(g, l,  512, 0);
    __builtin_amdgcn_global_load_async_to_lds_b128(g, l, 1024, 0);
    __builtin_amdgcn_global_load_async_to_lds_b128(g, l, 1536, 0);
    __builtin_amdgcn_global_load_async_to_lds_b128(g, l, 2048, 0);
    __builtin_amdgcn_global_load_async_to_lds_b128(g, l, 2560, 0);
    __builtin_amdgcn_global_load_async_to_lds_b128(g, l, 3072, 0);
    // tail: chunks 224..249 -> lanes 0..25 only (avoid reading past row/array)
    if (lane < 26)
        __builtin_amdgcn_global_load_async_to_lds_b128(g, l, 3584, 0);
}
#endif

// Per-lane scan of the row's float4 chunks (first-occurrence max semantics:
// strict '>' keeps the lowest index within a lane's increasing index order).
__device__ __forceinline__ void scan_f4(const float4* __restrict__ src, int lane,
                                        float& bv, int& bi) {
#pragma unroll
    for (int i = 0; i < 8; ++i) {
        int c = lane + 32 * i;
        if (c < ROW_F4_) {
            float4 v = src[c];
            int e = 4 * c;
            if (v.x > bv) { bv = v.x; bi = e; }
            if (v.y > bv) { bv = v.y; bi = e + 1; }
            if (v.z > bv) { bv = v.z; bi = e + 2; }
            if (v.w > bv) { bv = v.w; bi = e + 3; }
        }
    }
}

// Wave32 reduction (keep lowest index on ties == jnp.argmax), then lane 0
// bucketizes and accumulates the (class,bin) histogram with f32 atomics.
__device__ __forceinline__ void reduce_and_accum(float bv, int bi, int lane, long long r,
                                                 const int* __restrict__ labels,
                                                 float* __restrict__ ws) {
#pragma unroll
    for (int m = 16; m >= 1; m >>= 1) {
        float ov = __shfl_xor(bv, m, 32);
        int   oi = __shfl_xor(bi, m, 32);
        if (ov > bv || (ov == bv && oi < bi)) { bv = ov; bi = oi; }
    }
    if (lane == 0) {
        float conf = bv;
        // searchsorted(uppers, conf, 'left') == count of uppers strictly < conf,
        // clipped to N_BINS-1 (upper_15 = 1.0 can never be < conf for conf<=1).
        int bin = 0;
#pragma unroll
        for (int b = 1; b < N_BINS_; ++b)
            bin += (conf > (float)b * (1.0f / (float)N_BINS_)) ? 1 : 0;
        int seg = bi * N_BINS_ + bin;
        float cor = (bi == labels[r]) ? 1.0f : 0.0f;
        atomicAdd(ws + seg,             1.0f);
        atomicAdd(ws + NSEG_ + seg,     conf);
        atomicAdd(ws + 2 * NSEG_ + seg, cor);
    }
}

__global__ void zero_ws_kernel(float* __restrict__ ws, int n) {
    int i = blockIdx.x * blockDim.x + threadIdx.x;
    if (i < n) ws[i] = 0.0f;
}

__global__ __launch_bounds__(128) void sce_hist_kernel(const float* __restrict__ logits,
                                                       const int* __restrict__ labels,
                                                       float* __restrict__ ws, int nrows) {
    const int lane = threadIdx.x & 31;
    const int wid  = threadIdx.x >> 5;
    const long long nw = (long long)gridDim.x * 4;
    long long r = (long long)blockIdx.x * 4 + wid;

#if USE_ASYNC_LDS
    __shared__ float4 lbuf[4][2][256];  // 4 waves x double buffer x 4096B
    bool valid = (r < nrows);
    if (valid)
        async_prefetch_row(logits + r * (long long)N_CLASSES_, &lbuf[wid][0][0], lane);
    int pp = 0;
    while (valid) {
        long long rn = r + nw;
        bool vn = (rn < nrows);
        if (vn) {
            async_prefetch_row(logits + rn * (long long)N_CLASSES_, &lbuf[wid][pp ^ 1][0], lane);
            __builtin_amdgcn_s_wait_asynccnt(8);   // oldest 8 (row r) complete
        } else {
            __builtin_amdgcn_s_wait_asynccnt(0);   // drain last row
        }
        asm volatile("" ::: "memory");             // compiler fence for LDS reads
        float bv = -1.0f; int bi = 0;
        scan_f4(&lbuf[wid][pp][0], lane, bv, bi);
        reduce_and_accum(bv, bi, lane, r, labels, ws);
        pp ^= 1; r = rn; valid = vn;
    }
#else
    for (; r < nrows; r += nw) {
        const float4* rowp = (const float4*)(logits + r * (long long)N_CLASSES_);
        float bv = -1.0f; int bi = 0;
        scan_f4(rowp, lane, bv, bi);
        reduce_and_accum(bv, bi, lane, r, labels, ws);
    }
#endif
}

__global__ void sce_finalize_kernel(const float* __restrict__ ws, float* __restrict__ out,
                                    int nrows) {
    __shared__ float red[256];
    float acc = 0.0f;
    for (int s = threadIdx.x; s < NSEG_; s += 256) {
        float cnt = ws[s];
        if (cnt > 0.0f) {
            float cs = ws[NSEG_ + s];
            float rs = ws[2 * NSEG_ + s];
            float accb = rs / cnt;            // accuracy in bin
            float avgc = cs / cnt;            // avg confidence in bin
            float prop = cnt / (float)nrows;  // fraction of all predictions
            acc += fabsf(accb - avgc) * prop;
        }
    }
    red[threadIdx.x] = acc;
    __syncthreads();
#pragma unroll
    for (int off = 128; off > 0; off >>= 1) {
        if ((int)threadIdx.x < off) red[threadIdx.x] += red[threadIdx.x + off];
        __syncthreads();
    }
    if (threadIdx.x == 0) out[0] = red[0] / (float)N_CLASSES_;
}

extern "C" void kernel_launch(void* const* d_in, const int* in_sizes, int n_in,
                              void* d_out, int out_size, void* d_ws, size_t ws_size,
                              hipStream_t stream) {
    const float* logits = (const float*)d_in[0];
    const int*   labels = (const int*)d_in[1];
    float*       out    = (float*)d_out;
    float*       ws     = (float*)d_ws;      // 3 * 15000 f32 = 180 KB
    const int nrows = in_sizes[1];           // labels count == N rows

    zero_ws_kernel<<<(3 * NSEG_ + 255) / 256, 256, 0, stream>>>(ws, 3 * NSEG_);
    sce_hist_kernel<<<2048, 128, 0, stream>>>(logits, labels, ws, nrows);
    sce_finalize_kernel<<<1, 256, 0, stream>>>(ws, out, nrows);
}